// Patch_Embed_Rotate_64433099374841
// MI455X (gfx1250) — compile-verified
//
#include <hip/hip_runtime.h>
#include <hip/hip_bf16.h>
#include <stdint.h>

// Patch_Embed_Rotate: per-(batch,patch) conditional 16x16 spatial transpose.
// Pure bandwidth problem: 154 MB in + 154 MB out => ~13 us floor at 23.3 TB/s.
// Strategy: one block per horizontal PAIR of patches so every global access is a
// full 128B line; async global->LDS (ASYNCcnt, CDNA5) staging with TH_LOAD_NT
// (zero-reuse input, keep L2 clean); skewed LDS layout for conflict-free
// transposed gathers; 128-bit nontemporal stores out.

#define PATCH     16
#define HW        224
#define NP        14           // patches per side (224/16)
#define CHANNELS  3
#define LDS_PITCH 20           // floats/row: 16B-aligned float4 slots; 20k mod 64 distinct for k<16
#define TILE_F    (CHANNELS * PATCH * LDS_PITCH)   // 960 floats per patch tile
#define PP_STRIDE (TILE_F + 8)                     // +8-float skew: pp=1 shifted 8 banks

typedef float v4f __attribute__((ext_vector_type(4)));

__global__ __launch_bounds__(384) void patch_rotate_kernel(
    const float* __restrict__ x,
    const unsigned char* __restrict__ mask,   // jnp.bool_ = 1 byte per element
    float* __restrict__ out)
{
    __shared__ float tile[2 * PP_STRIDE];     // 7744 B

    const int bx = blockIdx.x;   // patch-pair column, 0..6  (wp = 2*bx + pp)
    const int hp = blockIdx.y;   // patch row, 0..13
    const int b  = blockIdx.z;   // batch

    const int t    = threadIdx.x;   // 0..383
    const int c    = t >> 7;        // channel 0..2
    const int q    = t & 127;
    const int row  = q >> 3;        // 0..15
    const int oct  = q & 7;         // 0..7 float4 slots across the 32-float pair row
    const int pp   = oct >> 2;      // which patch of the pair
    const int quad = oct & 3;       // float4 column group within the patch

    // Global offset: channel base (200704B), row stride (896B) and pair offset
    // (128B) are all multiples of 128B -> each wave issues whole 128B lines.
    const size_t chan_base = ((size_t)b * CHANNELS + c) * (size_t)(HW * HW);
    const size_t goff = chan_base
                      + (size_t)(hp * PATCH + row) * HW
                      + (size_t)(bx * 32 + oct * 4);

    // ---- async copy: global -> LDS, 16 bytes per lane, nontemporal ----
    float*   lptr   = tile + pp * PP_STRIDE + c * (PATCH * LDS_PITCH)
                           + row * LDS_PITCH + quad * 4;
    uint32_t ldsoff = (uint32_t)(size_t)lptr;  // flat LDS aperture: low 32 bits = LDS offset

    asm volatile("global_load_async_to_lds_b128 %0, %1, off th:TH_LOAD_NT"
                 :: "v"(ldsoff), "v"(x + goff)
                 : "memory");
    asm volatile("s_wait_asynccnt 0" ::: "memory");  // this wave's async writes landed
    __syncthreads();                                 // cross-wave visibility

    // Per-patch masks: two uniform scalar loads, per-lane select.
    const unsigned char* mrow = mask + (size_t)b * (NP * NP) + (size_t)(2 * bx) * NP + hp;
    const bool m0 = mrow[0]  != 0;
    const bool m1 = mrow[NP] != 0;
    const bool m  = pp ? m1 : m0;

    v4f v;
    if (m) {
        // transposed gather: column `row` from rows quad*4..quad*4+3
        // bank = 16*quad + 8*pp + row (mod 64) -> conflict-free across the wave
        const float* base = tile + pp * PP_STRIDE + c * (PATCH * LDS_PITCH);
        v.x = base[(quad * 4 + 0) * LDS_PITCH + row];
        v.y = base[(quad * 4 + 1) * LDS_PITCH + row];
        v.z = base[(quad * 4 + 2) * LDS_PITCH + row];
        v.w = base[(quad * 4 + 3) * LDS_PITCH + row];
    } else {
        v = *(const v4f*)lptr;    // one ds_load_b128 (16B-aligned)
    }

    // zero reuse -> nontemporal 128-bit store (full 128B lines per wave)
    __builtin_nontemporal_store(v, (v4f*)(out + goff));
}

extern "C" void kernel_launch(void* const* d_in, const int* in_sizes, int n_in,
                              void* d_out, int out_size, void* d_ws, size_t ws_size,
                              hipStream_t stream) {
    const float*         x    = (const float*)d_in[0];
    const unsigned char* mask = (const unsigned char*)d_in[1];
    float*               out  = (float*)d_out;

    const int B = in_sizes[0] / (CHANNELS * HW * HW);   // 256
    dim3 grid(NP / 2, NP, B);                           // (patch pair, patch row, batch)
    patch_rotate_kernel<<<grid, 384, 0, stream>>>(x, mask, out);
}